// PolicyID_73435350827514
// MI455X (gfx1250) — compile-verified
//
#include <hip/hip_runtime.h>
#include <math.h>

typedef float v2f __attribute__((ext_vector_type(2)));
typedef float v8f __attribute__((ext_vector_type(8)));
typedef int   v4i_vs __attribute__((vector_size(16)));   // matches builtin proto

#ifndef __has_builtin
#define __has_builtin(x) 0
#endif

// gfx1250 async global->LDS copy path (ASYNCcnt-tracked, no VGPR bounce).
// Probe (round 2 stderr): param type is 'int __vector(4) *' (generic ptr).
#if __has_builtin(__builtin_amdgcn_global_load_async_to_lds_b128)
#define USE_ASYNC_COPY 1
#define GLD_ASYNC_B128(gsrc, ldst)                                             \
    __builtin_amdgcn_global_load_async_to_lds_b128(                            \
        (v4i_vs*)(gsrc), (v4i_vs*)(ldst), 0, 0)
#else
#define USE_ASYNC_COPY 0
#endif

__device__ __forceinline__ void wait_async_all()
{
#if USE_ASYNC_COPY
#if __has_builtin(__builtin_amdgcn_s_wait_asynccnt)
    __builtin_amdgcn_s_wait_asynccnt(0);
#else
    asm volatile("s_wait_asynccnt 0x0" ::: "memory");
#endif
#endif
}

// ---------------------------------------------------------------------------
// GEMM: C = act(A[MxK] * B[KxN] + bias[N]), fp32, WMMA f32 16x16x4.
// 256 threads = 8 waves; tile 128(M) x 64(N), BK=16; double-buffered LDS with
// async global->LDS staging. Wave w owns rows [16w,16w+16) x all 64 cols.
// ---------------------------------------------------------------------------
#define TILE_M 128
#define TILE_N 64
#define TILE_K 16
#define THREADS 256

__device__ __forceinline__ void load_tiles(
    float (*__restrict__ As)[TILE_K], float (*__restrict__ Bs)[TILE_N + 4],
    const float* __restrict__ A, const float* __restrict__ B,
    int M, int N, int K, int mBase, int nBase, int k0, int tid)
{
    const bool full = (mBase + TILE_M <= M) && (nBase + TILE_N <= N) &&
                      (k0 + TILE_K <= K);
#if USE_ASYNC_COPY
    if (full) {
        // A tile: 128x16 = 512 B128 chunks (2/thread), 4 chunks per row.
        for (int j = 0; j < 2; ++j) {
            int ca = tid + j * THREADS;
            int r = ca >> 2, c = (ca & 3) * 4;
            GLD_ASYNC_B128(&A[(size_t)(mBase + r) * K + k0 + c], &As[r][c]);
        }
        // B tile: 16x64 = 256 B128 chunks (1/thread), 16 chunks per row.
        {
            int r = tid >> 4, c = (tid & 15) * 4;
            GLD_ASYNC_B128(&B[(size_t)(k0 + r) * N + nBase + c], &Bs[r][c]);
        }
        return;
    }
#endif
    for (int i = tid; i < TILE_M * TILE_K; i += THREADS) {
        int r = i / TILE_K, c = i % TILE_K;
        int gr = mBase + r, gc = k0 + c;
        As[r][c] = (gr < M && gc < K) ? A[(size_t)gr * K + gc] : 0.0f;
    }
    for (int i = tid; i < TILE_K * TILE_N; i += THREADS) {
        int r = i / TILE_N, c = i % TILE_N;
        int gr = k0 + r, gc = nBase + c;
        Bs[r][c] = (gr < K && gc < N) ? B[(size_t)gr * N + gc] : 0.0f;
    }
}

__global__ __launch_bounds__(THREADS)
void gemm_bias_act(const float* __restrict__ A, const float* __restrict__ B,
                   const float* __restrict__ bias, float* __restrict__ C,
                   int M, int N, int K, int relu)
{
    __shared__ __attribute__((aligned(16))) float As[2][TILE_M][TILE_K];
    __shared__ __attribute__((aligned(16))) float Bs[2][TILE_K][TILE_N + 4];

    const int tid  = threadIdx.x;
    const int lane = tid & 31;
    const int wave = tid >> 5;
    const int half = lane >> 4;
    const int l16  = lane & 15;

    const int mBase   = blockIdx.y * TILE_M;
    const int nBase   = blockIdx.x * TILE_N;
    const int waveRow = wave * 16;

    v8f acc[4];
    for (int t = 0; t < 4; ++t)
        for (int i = 0; i < 8; ++i) acc[t][i] = 0.0f;

    // Prologue: stage tile 0, then enter the ping-pong pipeline.
    int buf = 0;
    load_tiles(As[0], Bs[0], A, B, M, N, K, mBase, nBase, 0, tid);
    wait_async_all();
    __syncthreads();

    for (int k0 = 0; k0 < K; k0 += TILE_K) {
        const int kNext = k0 + TILE_K;
        if (kNext < K)
            load_tiles(As[buf ^ 1], Bs[buf ^ 1], A, B, M, N, K,
                       mBase, nBase, kNext, tid);
        // Prefetch the tile after next into L2 (global_prefetch_b8).
        if (kNext + TILE_K < K)
            __builtin_prefetch(&A[(size_t)(mBase + (tid & (TILE_M - 1))) * K +
                                  kNext + TILE_K], 0, 1);

        for (int kk = 0; kk < TILE_K; kk += 4) {
            // A frag 16x4: lane = row M, VGPR pair = K { 2*half, 2*half+1 }
            v2f a;
            a.x = As[buf][waveRow + l16][kk + 2 * half + 0];
            a.y = As[buf][waveRow + l16][kk + 2 * half + 1];
            for (int t = 0; t < 4; ++t) {
                // B frag 4x16: VGPR0 row K=2*half, VGPR1 row K=2*half+1
                v2f b;
                b.x = Bs[buf][kk + 2 * half + 0][t * 16 + l16];
                b.y = Bs[buf][kk + 2 * half + 1][t * 16 + l16];
                acc[t] = __builtin_amdgcn_wmma_f32_16x16x4_f32(
                    false, a, false, b, (short)0, acc[t], false, false);
            }
        }
        wait_async_all();
        __syncthreads();
        buf ^= 1;
    }

    // Epilogue: C/D layout -> VGPR i: M = i + 8*half, N = l16
    for (int t = 0; t < 4; ++t) {
        int col = nBase + t * 16 + l16;
        if (col < N) {
            float bv = bias[col];
            for (int i = 0; i < 8; ++i) {
                int row = mBase + waveRow + i + 8 * half;
                if (row < M) {
                    float v = acc[t][i] + bv;
                    if (relu) v = fmaxf(v, 0.0f);
                    C[(size_t)row * N + col] = v;
                }
            }
        }
    }
}

// ---------------------------------------------------------------------------
// Constraint projection: one workgroup (256 thr = 8 waves) per batch item.
//   v   = relu(A r - b)                (192)
//   G   = A^T A + reg I                (96x96, WMMA, 6x6 tiles, K=192)
//   y   = G^{-1} (A^T v)               (SPD Gaussian elim + back-sub in LDS)
//   out = lb + (ub-lb)*sigmoid(r - y)
// ---------------------------------------------------------------------------
#define CM 192
#define CN 96

__global__ __launch_bounds__(256)
void constraint_kernel(const float* __restrict__ R, const float* __restrict__ A,
                       const float* __restrict__ bvec, const float* __restrict__ lb,
                       const float* __restrict__ ub, float* __restrict__ out,
                       float reg)
{
    __shared__ __attribute__((aligned(16))) float As[CM * CN]; // 72 KB
    __shared__ float Gm[CN * CN];                              // 36 KB
    __shared__ float vv[CM];
    __shared__ float bb[CM];
    __shared__ float rr[CN];
    __shared__ float rhs[CN];

    const int bidx = blockIdx.x;
    const int tid  = threadIdx.x;
    const int lane = tid & 31;
    const int wave = tid >> 5;
    const int half = lane >> 4;
    const int l16  = lane & 15;

    const float* Ab = A + (size_t)bidx * CM * CN;
#if USE_ASYNC_COPY
    // Contiguous 72 KB image: 4608 B128 chunks, 18 per thread, async to LDS.
    for (int c = tid; c < (CM * CN) / 4; c += 256)
        GLD_ASYNC_B128(Ab + (size_t)c * 4, &As[c * 4]);
#else
    for (int i = tid; i < CM * CN; i += 256) As[i] = Ab[i];
#endif
    if (tid < CN) rr[tid] = R[(size_t)bidx * CN + tid];
    if (tid < CM) bb[tid] = bvec[(size_t)bidx * CM + tid];
    wait_async_all();
    __syncthreads();

    // violation = relu(A r - b)
    if (tid < CM) {
        float s = 0.0f;
        for (int n = 0; n < CN; ++n) s += As[tid * CN + n] * rr[n];
        vv[tid] = fmaxf(s - bb[tid], 0.0f);
    }

    // G = A^T A + reg I via WMMA: 36 tiles of 16x16, K over 192 in steps of 4.
    // A-operand = (A^T) tile (rows n, K = m); B-operand = A tile (rows m, cols k).
    for (int t = wave; t < 36; t += 8) {
        int ti = t / 6, tj = t % 6;
        v8f acc;
        for (int i = 0; i < 8; ++i) acc[i] = 0.0f;
        int n    = ti * 16 + l16;
        int kcol = tj * 16 + l16;
        for (int mk = 0; mk < CM; mk += 4) {
            int m0 = mk + 2 * half;
            v2f a; a.x = As[m0 * CN + n];    a.y = As[(m0 + 1) * CN + n];
            v2f b; b.x = As[m0 * CN + kcol]; b.y = As[(m0 + 1) * CN + kcol];
            acc = __builtin_amdgcn_wmma_f32_16x16x4_f32(
                false, a, false, b, (short)0, acc, false, false);
        }
        for (int i = 0; i < 8; ++i) {
            int row = ti * 16 + i + 8 * half;
            int col = tj * 16 + l16;
            float g = acc[i];
            if (row == col) g += reg;
            Gm[row * CN + col] = g;
        }
    }
    __syncthreads();

    // rhs = A^T v
    if (tid < CN) {
        float s = 0.0f;
        for (int m = 0; m < CM; ++m) s += As[m * CN + tid] * vv[m];
        rhs[tid] = s;
    }
    __syncthreads();

    // Forward elimination (G is SPD + reg*I: no pivoting). U gets unit diag.
    for (int k = 0; k < CN; ++k) {
        float piv = Gm[k * CN + k];
        __syncthreads();
        float inv = 1.0f / piv;
        for (int j = k + tid; j < CN; j += 256) Gm[k * CN + j] *= inv;
        if (tid == 0) rhs[k] *= inv;
        __syncthreads();
        for (int i = k + 1 + tid; i < CN; i += 256) {
            float f  = Gm[i * CN + k];
            float rk = rhs[k];
            for (int j = k + 1; j < CN; ++j) Gm[i * CN + j] -= f * Gm[k * CN + j];
            rhs[i] -= f * rk;
        }
        __syncthreads();
    }

    // Back substitution (column sweep; rhs becomes y in place).
    for (int i = CN - 1; i >= 0; --i) {
        float yi = rhs[i];
        for (int j = tid; j < i; j += 256) rhs[j] -= Gm[j * CN + i] * yi;
        __syncthreads();
    }

    // out = lb + (ub - lb) * sigmoid(r - y)
    if (tid < CN) {
        float c = rr[tid] - rhs[tid];
        float s = 1.0f / (1.0f + expf(-c));
        float l = lb[(size_t)bidx * CN + tid];
        float u = ub[(size_t)bidx * CN + tid];
        out[(size_t)bidx * CN + tid] = l + (u - l) * s;
    }
}

// ---------------------------------------------------------------------------
extern "C" void kernel_launch(void* const* d_in, const int* in_sizes, int n_in,
                              void* d_out, int out_size, void* d_ws, size_t ws_size,
                              hipStream_t stream)
{
    const float* state = (const float*)d_in[0];
    const float* W1    = (const float*)d_in[1];
    const float* bh1   = (const float*)d_in[2];
    const float* W2    = (const float*)d_in[3];
    const float* bh2   = (const float*)d_in[4];
    const float* W3    = (const float*)d_in[5];
    const float* bh3   = (const float*)d_in[6];
    const float* A     = (const float*)d_in[7];
    const float* bvec  = (const float*)d_in[8];
    const float* lb    = (const float*)d_in[9];
    const float* ub    = (const float*)d_in[10];

    const int H = in_sizes[2];              // 1024
    const int S = in_sizes[1] / H;          // 512
    const int B = in_sizes[0] / S;          // 8192
    const int N = in_sizes[6];              // 96  (== CN)
    (void)n_in; (void)out_size; (void)ws_size;

    float* X1 = (float*)d_ws;               // B*H fp32
    float* X2 = X1 + (size_t)B * H;         // B*H fp32
    float* R  = X1;                         // raw actions reuse X1 (B*N <= B*H)

    dim3 blk(THREADS);
    dim3 g1((H + TILE_N - 1) / TILE_N, (B + TILE_M - 1) / TILE_M);
    gemm_bias_act<<<g1, blk, 0, stream>>>(state, W1, bh1, X1, B, H, S, 1);

    dim3 g2((H + TILE_N - 1) / TILE_N, (B + TILE_M - 1) / TILE_M);
    gemm_bias_act<<<g2, blk, 0, stream>>>(X1, W2, bh2, X2, B, H, H, 1);

    dim3 g3((N + TILE_N - 1) / TILE_N, (B + TILE_M - 1) / TILE_M);
    gemm_bias_act<<<g3, blk, 0, stream>>>(X2, W3, bh3, R, B, N, H, 0);

    constraint_kernel<<<dim3(B), blk, 0, stream>>>(R, A, bvec, lb, ub,
                                                   (float*)d_out, 0.01f);
}